// LSTM_10651518894911
// MI455X (gfx1250) — compile-verified
//
#include <hip/hip_runtime.h>
#include <math.h>

// ---- problem dims (match reference) ----
#define T_LEN   8192
#define IN_K    512
#define HID     2048
#define FOURH   8192
#define OUT_N   512

// ---- recurrence partitioning ----
#define NB      128           // persistent blocks (co-resident: ~9KB LDS, 256 thr each)
#define UPB     (HID / NB)    // 16 hidden units per block
#define ROWS_PB (4 * UPB)     // 64 gate rows per block

typedef __attribute__((ext_vector_type(2))) float v2f;
typedef __attribute__((ext_vector_type(8))) float v8f;
typedef __attribute__((ext_vector_type(4))) unsigned v4u;
typedef __attribute__((ext_vector_type(8))) int     v8i_t;
typedef __attribute__((ext_vector_type(4))) int     v4i_t;

// =====================================================================
// Kernel 1: x_proj[T,4H] = X[T,K] @ Wih[4H,K]^T + (b_ih + b_hh)
// One wave computes a 16x16 f32 tile via V_WMMA_F32_16X16X4_F32.
// A frag (16x4):  lane L, reg v -> M = L%16, K = (L/16)*2 + v
// B frag (4x16):  lane L, reg v -> N = L%16, K = (L/16)*2 + v  (B[k][n]=Wih[n][k])
// C/D  (16x16):   lane L, reg r -> M = r + (L/16)*8, N = L%16
// =====================================================================
__global__ __launch_bounds__(256) void gemm_xproj_wmma(
    const float* __restrict__ X, const float* __restrict__ Wih,
    const float* __restrict__ bih, const float* __restrict__ bhh,
    float* __restrict__ xproj)
{
    const int lane = threadIdx.x & 31;
    const int wave = threadIdx.x >> 5;
    const int m0   = blockIdx.y * 16;
    const int n0   = blockIdx.x * 128 + wave * 16;

    const int mrow  = m0 + (lane & 15);
    const int nrow  = n0 + (lane & 15);
    const int khalf = (lane >> 4) * 2;

    const float* xptr = X   + (size_t)mrow * IN_K + khalf;
    const float* wptr = Wih + (size_t)nrow * IN_K + khalf;

    v8f c = {};
#pragma unroll 8
    for (int k0 = 0; k0 < IN_K; k0 += 4) {
        v2f a = *(const v2f*)(xptr + k0);
        v2f b = *(const v2f*)(wptr + k0);
        c = __builtin_amdgcn_wmma_f32_16x16x4_f32(
                false, a, false, b, (short)0, c, false, false);
    }

    const int   ncol  = n0 + (lane & 15);
    const float bias  = bih[ncol] + bhh[ncol];
    const int   mbase = m0 + (lane >> 4) * 8;
#pragma unroll
    for (int r = 0; r < 8; ++r) {
        xproj[(size_t)(mbase + r) * FOURH + ncol] = c[r] + bias;
    }
}

// =====================================================================
// Kernel 0: zero the h vector and the per-step barrier counters
// =====================================================================
__global__ void init_state(float* __restrict__ h_g, unsigned* __restrict__ cnt)
{
    int tid = blockIdx.x * blockDim.x + threadIdx.x;
    if (tid < HID)   h_g[tid] = 0.0f;
    if (tid < T_LEN) cnt[tid] = 0u;
}

__device__ __forceinline__ float sigmoidf_(float x) {
    return 1.0f / (1.0f + __expf(-x));
}

// =====================================================================
// Kernel 2: persistent recurrence. NB blocks, block b owns hidden units
// [b*UPB, (b+1)*UPB). w_hh (64 MB fp32) lives in the 192 MB L2 and is
// re-streamed every step (64 MB x 8192 steps stays on-die). h (8 KB) is
// staged into LDS each step by the Tensor Data Mover; the block waits on
// TENSORcnt. Device-scope split barrier per step: release atomicAdd on a
// per-step counter + acquire spin (no counter reset needed).
// =====================================================================
__global__ __launch_bounds__(256) void lstm_recurrence(
    const float* __restrict__ xproj, const float* __restrict__ whh,
    float* __restrict__ h_g, unsigned* __restrict__ cnt)
{
    __shared__ float h_s[HID];        // full hidden state (8 KB), TDM destination
    __shared__ float gates_s[ROWS_PB];
    __shared__ float c_s[UPB];        // persistent cell state for this block

    const int b    = blockIdx.x;
    const int tid  = threadIdx.x;
    const int wave = tid >> 5;
    const int lane = tid & 31;

    if (tid < UPB) c_s[tid] = 0.0f;

    // ---- build the TDM descriptor (D#) once; identical every step ----
    // Group 0: [1:0]=count=1 | [63:32]=lds_addr | [120:64]=global_addr | [127:126]=type=2
    const unsigned           lds_off = (unsigned)(uintptr_t)(&h_s[0]);
    const unsigned long long gaddr   = (unsigned long long)(uintptr_t)h_g;
    v4u g0 = {};
    g0[0] = 1u;                                        // count=1 (valid, user mode)
    g0[1] = lds_off;                                   // lds_addr (bytes)
    g0[2] = (unsigned)(gaddr & 0xFFFFFFFFull);         // global_addr[31:0]
    g0[3] = (unsigned)((gaddr >> 32) & 0x1FFFFFFull)   // global_addr[56:32]
          | (2u << 30);                                // type = 2 ("image")
    // Group 1: wg_mask=0 | data_size=2(4B) | tensor_dim0=2048 | tensor_dim1=1
    //          tile_dim0=2048 | tile_dim1=1 | tile_dim2=0 | dim0_stride=2048
    v8i_t g1 = {};
    g1[0] = (int)(2u << 16);        // [17:16] data_size = 4 bytes
    g1[1] = (int)(2048u << 16);     // bits[63:48]  = tensor_dim0[15:0]
    g1[2] = (int)(1u << 16);        // bits[79:64]=dim0 hi=0; bits[95:80]=tensor_dim1=1
    g1[3] = (int)(2048u << 16);     // bits[111:96]=dim1 hi=0; bits[127:112]=tile_dim0
    g1[4] = 1;                      // bits[143:128]=tile_dim1=1; tile_dim2=0
    g1[5] = 2048;                   // tensor_dim0_stride[31:0]
    v4i_t gz4 = {};                 // groups 2/3 unused (<=2D tile)
    v8i_t gz8 = {};                 // trailing zero group (6-arg builtin form)

    __syncthreads();

    for (int t = 0; t < T_LEN; ++t) {
        // stage full h into LDS via the Tensor Data Mover (wave 0 issues;
        // descriptor is wave-uniform -> SGPRs; TDM ignores EXEC)
        if (wave == 0) {
            __builtin_amdgcn_tensor_load_to_lds(g0, g1, gz4, gz4, gz8, 0);
            __builtin_amdgcn_s_wait_tensorcnt(0);
        }
        __syncthreads();

        const float* xp = xproj + (size_t)t * FOURH;
        const float4* h4 = (const float4*)h_s;

        // 64 gate rows per block; each wave handles 8, lanes split K by float4.
#pragma unroll 2
        for (int pi = 0; pi < 8; ++pi) {
            const int p   = wave * 8 + pi;      // p = g*UPB + u
            const int g   = p >> 4;
            const int u   = p & (UPB - 1);
            const int row = g * HID + b * UPB + u;
            const float4* w4 = (const float4*)(whh + (size_t)row * HID);
            float acc = 0.0f;
#pragma unroll 4
            for (int k = lane; k < HID / 4; k += 32) {   // 16 b128 loads/lane
                const float4 w = w4[k];
                const float4 h = h4[k];
                acc += w.x * h.x + w.y * h.y + w.z * h.z + w.w * h.w;
            }
#pragma unroll
            for (int off = 16; off > 0; off >>= 1)
                acc += __shfl_xor(acc, off, 32);
            if (lane == 0) gates_s[p] = acc + xp[row];
        }
        __syncthreads();

        // pointwise LSTM cell update for this block's 16 units
        if (tid < UPB) {
            const int u = tid;
            const float ig = sigmoidf_(gates_s[0 * UPB + u]);
            const float fg = sigmoidf_(gates_s[1 * UPB + u]);
            const float gg = tanhf    (gates_s[2 * UPB + u]);
            const float og = sigmoidf_(gates_s[3 * UPB + u]);
            const float cn = fg * c_s[u] + ig * gg;
            c_s[u] = cn;
            h_g[b * UPB + u] = og * tanhf(cn);
        }
        __syncthreads();   // all h writes of this block issued

        // device-scope split barrier on per-step counter
        if (tid == 0) {
            __hip_atomic_fetch_add(&cnt[t], 1u, __ATOMIC_RELEASE,
                                   __HIP_MEMORY_SCOPE_AGENT);
            while (__hip_atomic_load(&cnt[t], __ATOMIC_ACQUIRE,
                                     __HIP_MEMORY_SCOPE_AGENT) < (unsigned)NB) {
                __builtin_amdgcn_s_sleep(1);
            }
        }
        __syncthreads();
    }
}

// =====================================================================
// Kernel 3: out[512] = hT @ w_out.T + b_out   (tiny; wave-per-output)
// =====================================================================
__global__ __launch_bounds__(256) void out_proj(
    const float* __restrict__ h_g, const float* __restrict__ wout,
    const float* __restrict__ bout, float* __restrict__ out)
{
    const int wave = threadIdx.x >> 5;
    const int lane = threadIdx.x & 31;
    const int o    = blockIdx.x * 8 + wave;    // 64 blocks -> 512 outputs
    const float4* w4 = (const float4*)(wout + (size_t)o * HID);
    const float4* h4 = (const float4*)h_g;
    float acc = 0.0f;
#pragma unroll 4
    for (int k = lane; k < HID / 4; k += 32) {
        const float4 w = w4[k];
        const float4 h = h4[k];
        acc += w.x * h.x + w.y * h.y + w.z * h.z + w.w * h.w;
    }
#pragma unroll
    for (int off = 16; off > 0; off >>= 1)
        acc += __shfl_xor(acc, off, 32);
    if (lane == 0) out[o] = acc + bout[o];
}

// =====================================================================
extern "C" void kernel_launch(void* const* d_in, const int* in_sizes, int n_in,
                              void* d_out, int out_size, void* d_ws, size_t ws_size,
                              hipStream_t stream)
{
    const float* x    = (const float*)d_in[0];  // [8192, 512]
    const float* wih  = (const float*)d_in[1];  // [8192, 512]
    const float* whh  = (const float*)d_in[2];  // [8192, 2048]
    const float* bih  = (const float*)d_in[3];  // [8192]
    const float* bhh  = (const float*)d_in[4];  // [8192]
    const float* wout = (const float*)d_in[5];  // [512, 2048]
    const float* bout = (const float*)d_in[6];  // [512]
    float* out = (float*)d_out;                 // [512]

    // workspace layout
    char* ws = (char*)d_ws;
    float*    xproj = (float*)ws;                                   // 256 MB
    size_t    off   = (size_t)T_LEN * FOURH * sizeof(float);
    float*    h_g   = (float*)(ws + off);                           // 8 KB
    off += (size_t)HID * sizeof(float);
    unsigned* cnt   = (unsigned*)(ws + off);                        // 32 KB

    // 0) zero h and barrier counters (ws is poisoned / left dirty)
    init_state<<<dim3((T_LEN + 255) / 256), dim3(256), 0, stream>>>(h_g, cnt);

    // 1) x_proj GEMM via f32 WMMA: grid (4H/128, T/16)
    gemm_xproj_wmma<<<dim3(FOURH / 128, T_LEN / 16), dim3(256), 0, stream>>>(
        x, wih, bih, bhh, xproj);

    // 2) persistent recurrence over 8192 steps
    lstm_recurrence<<<dim3(NB), dim3(256), 0, stream>>>(xproj, whh, h_g, cnt);

    // 3) final projection
    out_proj<<<dim3(OUT_N / 8), dim3(256), 0, stream>>>(h_g, wout, bout, out);
}